// GAT_2156073582616
// MI455X (gfx1250) — compile-verified
//
#include <hip/hip_runtime.h>
#include <math.h>

#define NNODES 50000
#define NEDGES 800000
#define DIN    256
#define NHEAD  8
#define NCHAN  32
#define DHC    256
#define NCLS   40
#define SLOPE  0.2f
#define LNEPS  1e-5f

typedef float v2f __attribute__((ext_vector_type(2)));
typedef float v8f __attribute__((ext_vector_type(8)));

// ---------------------------------------------------------------------------
// fp32 WMMA GEMM:  C[M x Ncol] = A[M x K] @ B[K x Ncol]   (all row-major)
// blockDim = 256 (8 waves); each wave computes a 32x32 macro-tile of C as a
// 2x2 grid of 16x16 WMMA tiles -> per k-step: 2 A-frags + 2 B-frags, 4 WMMAs.
// grid.x = Ncol/32 ; grid.y = ceil(ceil(M/32) / 8)
// ---------------------------------------------------------------------------
__global__ void gat_gemm_wmma_f32(const float* __restrict__ A,
                                  const float* __restrict__ B,
                                  float* __restrict__ C,
                                  int M, int K, int Ncol) {
    const int wave = threadIdx.x >> 5;
    const int lane = threadIdx.x & 31;
    const int nMacroRows = (M + 31) >> 5;
    const int macroRow = blockIdx.y * 8 + wave;
    if (macroRow >= nMacroRows) return;       // wave-uniform guard (EXEC stays full)
    const int row0 = macroRow * 32;
    const int col0 = blockIdx.x * 32;
    const int m  = lane & 15;                  // row (A) / col (B,C) within tile
    const int hh = lane >> 4;                  // half-wave selects K pair / C row half
    const bool tile1Valid = (row0 + 32) <= M;  // wave-uniform

    // clamp second row-tile loads so EXEC can stay all-ones for WMMA
    int r1 = row0 + 16 + m;
    if (r1 > M - 1) r1 = M - 1;

    const float* __restrict__ a0p = A + (size_t)(row0 + m) * K + 2 * hh;
    const float* __restrict__ a1p = A + (size_t)r1 * K + 2 * hh;
    const float* __restrict__ bp  = B + (size_t)(2 * hh) * Ncol + col0 + m;

    v8f c00 = {}, c01 = {}, c10 = {}, c11 = {};
    #pragma unroll 2
    for (int k0 = 0; k0 < K; k0 += 4) {
        v2f a0, a1, b0, b1;
        a0.x = a0p[0];      a0.y = a0p[1];
        a1.x = a1p[0];      a1.y = a1p[1];
        b0.x = bp[0];       b0.y = bp[Ncol];
        b1.x = bp[16];      b1.y = bp[Ncol + 16];
        c00 = __builtin_amdgcn_wmma_f32_16x16x4_f32(false, a0, false, b0,
                                                    (short)0, c00, false, false);
        c01 = __builtin_amdgcn_wmma_f32_16x16x4_f32(false, a0, false, b1,
                                                    (short)0, c01, false, false);
        c10 = __builtin_amdgcn_wmma_f32_16x16x4_f32(false, a1, false, b0,
                                                    (short)0, c10, false, false);
        c11 = __builtin_amdgcn_wmma_f32_16x16x4_f32(false, a1, false, b1,
                                                    (short)0, c11, false, false);
        a0p += 4;
        a1p += 4;
        bp  += 4 * (size_t)Ncol;
    }

    // C/D layout: VGPR r -> row (r + 8*hh), col = lane&15
    float* __restrict__ crow0 = C + (size_t)(row0 + hh * 8) * Ncol + col0 + m;
    #pragma unroll
    for (int r = 0; r < 8; ++r) {
        crow0[(size_t)r * Ncol]      = c00[r];
        crow0[(size_t)r * Ncol + 16] = c01[r];
    }
    if (tile1Valid) {                          // wave-uniform branch
        float* __restrict__ crow1 = crow0 + (size_t)16 * Ncol;
        #pragma unroll
        for (int r = 0; r < 8; ++r) {
            crow1[(size_t)r * Ncol]      = c10[r];
            crow1[(size_t)r * Ncol + 16] = c11[r];
        }
    }
}

// ---------------------------------------------------------------------------
// alpha_src[n,h] = sum_c h[n,h,c]*a_s[h,c] ; alpha_dst analogously
// ---------------------------------------------------------------------------
__global__ void gat_alpha_kernel(const float* __restrict__ hfeat,
                                 const float* __restrict__ a_s,
                                 const float* __restrict__ a_d,
                                 float* __restrict__ asrc,
                                 float* __restrict__ adst) {
    int idx = blockIdx.x * blockDim.x + threadIdx.x;   // over N*H
    if (idx >= NNODES * NHEAD) return;
    const int n  = idx >> 3;
    const int hd = idx & 7;
    const float* __restrict__ hp = hfeat + (size_t)n * DHC + hd * NCHAN;
    const float* __restrict__ ws = a_s + hd * NCHAN;
    const float* __restrict__ wd = a_d + hd * NCHAN;
    float ss = 0.f, sd = 0.f;
    #pragma unroll 8
    for (int c = 0; c < NCHAN; ++c) {
        const float hv = hp[c];
        ss += hv * ws[c];
        sd += hv * wd[c];
    }
    asrc[idx] = ss;
    adst[idx] = sd;
}

__global__ void gat_init_mz(float* __restrict__ m, float* __restrict__ z, int n) {
    int idx = blockIdx.x * blockDim.x + threadIdx.x;
    if (idx < n) { m[idx] = -INFINITY; z[idx] = 0.f; }
}

__device__ __forceinline__ void atomicMaxFloat(float* addr, float val) {
    if (val >= 0.f) atomicMax((int*)addr, __float_as_int(val));
    else            atomicMin((unsigned int*)addr, __float_as_uint(val));
}

__device__ __forceinline__ float leaky(float v) {
    return v > 0.f ? v : SLOPE * v;
}

__device__ __forceinline__ void edge_sd(const int* __restrict__ ei, int e,
                                        int& s, int& d) {
    if (e < NEDGES) { s = ei[e]; d = ei[NEDGES + e]; }
    else            { s = d = e - NEDGES; }
}

// pass 1: segment max over incoming edges
__global__ void gat_edge_max(const int* __restrict__ ei,
                             const float* __restrict__ asrc,
                             const float* __restrict__ adst,
                             float* __restrict__ mbuf) {
    int idx = blockIdx.x * blockDim.x + threadIdx.x;   // over Eprime*H
    const int tot = (NEDGES + NNODES) * NHEAD;
    if (idx >= tot) return;
    const int e = idx >> 3, hd = idx & 7;
    int s, d; edge_sd(ei, e, s, d);
    const float v = leaky(asrc[s * NHEAD + hd] + adst[d * NHEAD + hd]);
    atomicMaxFloat(&mbuf[d * NHEAD + hd], v);
}

// pass 2: segment sum of exp(e - m)
__global__ void gat_edge_sum(const int* __restrict__ ei,
                             const float* __restrict__ asrc,
                             const float* __restrict__ adst,
                             const float* __restrict__ mbuf,
                             float* __restrict__ zbuf) {
    int idx = blockIdx.x * blockDim.x + threadIdx.x;
    const int tot = (NEDGES + NNODES) * NHEAD;
    if (idx >= tot) return;
    const int e = idx >> 3, hd = idx & 7;
    int s, d; edge_sd(ei, e, s, d);
    const float v = leaky(asrc[s * NHEAD + hd] + adst[d * NHEAD + hd]);
    atomicAdd(&zbuf[d * NHEAD + hd], expf(v - mbuf[d * NHEAD + hd]));
}

// pass 3: agg[dst] += h[src] * alpha  — one wave per edge, 8 channels/lane
__global__ void gat_edge_agg(const int* __restrict__ ei,
                             const float* __restrict__ asrc,
                             const float* __restrict__ adst,
                             const float* __restrict__ mbuf,
                             const float* __restrict__ zbuf,
                             const float* __restrict__ hfeat,
                             float* __restrict__ agg) {
    const int gwave = (int)((blockIdx.x * blockDim.x + threadIdx.x) >> 5);
    const int lane  = threadIdx.x & 31;
    if (gwave >= NEDGES + NNODES) return;              // wave-uniform guard
    int s, d; edge_sd(ei, gwave, s, d);
    const int hd = lane >> 2;                           // head of channels [lane*8, +8)
    const float v = leaky(asrc[s * NHEAD + hd] + adst[d * NHEAD + hd]);
    const float alpha = expf(v - mbuf[d * NHEAD + hd]) / zbuf[d * NHEAD + hd];
    const float4* __restrict__ hs =
        (const float4*)(hfeat + (size_t)s * DHC + lane * 8);
    const float4 h0 = hs[0], h1 = hs[1];
    float* __restrict__ ag = agg + (size_t)d * DHC + lane * 8;
    atomicAdd(ag + 0, h0.x * alpha);
    atomicAdd(ag + 1, h0.y * alpha);
    atomicAdd(ag + 2, h0.z * alpha);
    atomicAdd(ag + 3, h0.w * alpha);
    atomicAdd(ag + 4, h1.x * alpha);
    atomicAdd(ag + 5, h1.y * alpha);
    atomicAdd(ag + 6, h1.z * alpha);
    atomicAdd(ag + 7, h1.w * alpha);
}

// bias + LayerNorm + ReLU, in place. One wave per node, 8 channels per lane.
__global__ void gat_ln_relu(float* __restrict__ buf,
                            const float* __restrict__ bias,
                            const float* __restrict__ g,
                            const float* __restrict__ be) {
    const int wv   = (int)((blockIdx.x * blockDim.x + threadIdx.x) >> 5);
    const int lane = threadIdx.x & 31;
    if (wv >= NNODES) return;                          // wave-uniform guard
    float* __restrict__ row = buf + (size_t)wv * DHC;
    float vals[8];
    float s = 0.f;
    #pragma unroll
    for (int j = 0; j < 8; ++j) {
        const int c = lane + 32 * j;
        vals[j] = row[c] + bias[c];
        s += vals[j];
    }
    #pragma unroll
    for (int off = 16; off > 0; off >>= 1) s += __shfl_xor(s, off, 32);
    const float mu = s * (1.f / 256.f);
    float var = 0.f;
    #pragma unroll
    for (int j = 0; j < 8; ++j) {
        const float dv = vals[j] - mu;
        var += dv * dv;
    }
    #pragma unroll
    for (int off = 16; off > 0; off >>= 1) var += __shfl_xor(var, off, 32);
    const float inv = rsqrtf(var * (1.f / 256.f) + LNEPS);
    #pragma unroll
    for (int j = 0; j < 8; ++j) {
        const int c = lane + 32 * j;
        const float y = (vals[j] - mu) * inv * g[c] + be[c];
        row[c] = fmaxf(y, 0.f);
    }
}

// final classifier: out[n,c] = bout[c] + sum_k h[n,k]*Wout[k,c]
__global__ void gat_out_gemm(const float* __restrict__ hfeat,
                             const float* __restrict__ Wout,
                             const float* __restrict__ bout,
                             float* __restrict__ out) {
    int idx = blockIdx.x * blockDim.x + threadIdx.x;   // over N*NCLS
    if (idx >= NNODES * NCLS) return;
    const int n = idx / NCLS, c = idx % NCLS;
    const float* __restrict__ hp = hfeat + (size_t)n * DHC;
    float s = bout[c];
    #pragma unroll 8
    for (int k = 0; k < DHC; ++k)
        s += hp[k] * Wout[k * NCLS + c];
    out[idx] = s;
}

// ---------------------------------------------------------------------------
extern "C" void kernel_launch(void* const* d_in, const int* in_sizes, int n_in,
                              void* d_out, int out_size, void* d_ws, size_t ws_size,
                              hipStream_t stream) {
    const float* x    = (const float*)d_in[0];
    const int*   ei   = (const int*)d_in[1];
    const float* W[2]  = {(const float*)d_in[2], (const float*)d_in[8]};
    const float* As[2] = {(const float*)d_in[3], (const float*)d_in[9]};
    const float* Ad[2] = {(const float*)d_in[4], (const float*)d_in[10]};
    const float* Bb[2] = {(const float*)d_in[5], (const float*)d_in[11]};
    const float* Gg[2] = {(const float*)d_in[6], (const float*)d_in[12]};
    const float* Be[2] = {(const float*)d_in[7], (const float*)d_in[13]};
    const float* Wout = (const float*)d_in[14];
    const float* bout = (const float*)d_in[15];
    float* out = (float*)d_out;

    char* ws = (char*)d_ws;
    const size_t featBytes = (size_t)NNODES * DHC * sizeof(float);   // 51.2 MB
    float* bufA = (float*)ws;                       // h = linear(x)
    float* bufB = (float*)(ws + featBytes);         // aggregation / normalized feats
    float* asrc = (float*)(ws + 2 * featBytes);     // N*H
    float* adst = asrc + NNODES * NHEAD;
    float* mbuf = adst + NNODES * NHEAD;
    float* zbuf = mbuf + NNODES * NHEAD;

    const int EP     = NEDGES + NNODES;
    const int NH_TOT = NNODES * NHEAD;

    const int nMacroRows = (NNODES + 31) / 32;          // 1563
    const dim3 ggrid(DHC / 32, (nMacroRows + 7) / 8);   // 8 x 196

    const float* cur = x;
    for (int L = 0; L < 2; ++L) {
        // h = cur @ W[L]   (WMMA fp32, 2x2 register blocking)
        gat_gemm_wmma_f32<<<ggrid, 256, 0, stream>>>(cur, W[L], bufA,
                                                     NNODES, DHC, DHC);
        gat_alpha_kernel<<<(NH_TOT + 255) / 256, 256, 0, stream>>>(
            bufA, As[L], Ad[L], asrc, adst);
        gat_init_mz<<<(NH_TOT + 255) / 256, 256, 0, stream>>>(mbuf, zbuf, NH_TOT);
        hipMemsetAsync(bufB, 0, featBytes, stream);
        gat_edge_max<<<(EP * NHEAD + 255) / 256, 256, 0, stream>>>(
            ei, asrc, adst, mbuf);
        gat_edge_sum<<<(EP * NHEAD + 255) / 256, 256, 0, stream>>>(
            ei, asrc, adst, mbuf, zbuf);
        gat_edge_agg<<<(EP + 7) / 8, 256, 0, stream>>>(
            ei, asrc, adst, mbuf, zbuf, bufA, bufB);
        gat_ln_relu<<<(NNODES + 7) / 8, 256, 0, stream>>>(
            bufB, Bb[L], Gg[L], Be[L]);
        cur = bufB;
    }
    gat_out_gemm<<<(NNODES * NCLS + 255) / 256, 256, 0, stream>>>(
        bufB, Wout, bout, out);
}